// HieraGConv_59562606461631
// MI455X (gfx1250) — compile-verified
//
#include <hip/hip_runtime.h>
#include <math.h>

typedef __attribute__((ext_vector_type(2))) float v2f;
typedef __attribute__((ext_vector_type(8))) float v8f;

#define DH 128            // feature width everywhere (D_IN = D_HID = D_OUT = 128)
#define TB 256            // default block size

// ---------------------------------------------------------------- utilities
__global__ void k_zero_f32(float* __restrict__ p, long n) {
    long i = (long)blockIdx.x * blockDim.x + threadIdx.x;
    if (i < n) p[i] = 0.0f;
}

__global__ void k_neg_copy(const float* __restrict__ x, float* __restrict__ y, long n) {
    long i = (long)blockIdx.x * blockDim.x + threadIdx.x;
    if (i < n) y[i] = -x[i];
}

// d_out[:, f] += bias[f]
__global__ void k_add_row_bias(float* __restrict__ out, const float* __restrict__ bias, long n) {
    long i = (long)blockIdx.x * blockDim.x + threadIdx.x;
    if (i < n) out[i] += bias[(int)(i & (DH - 1))];
}

// ---------------------------------------------------------------- histograms
__global__ void k_deg_accum(const int* __restrict__ row, float* __restrict__ deg, int nE) {
    int e = blockIdx.x * blockDim.x + threadIdx.x;
    if (e < nE) atomicAdd(&deg[row[e]], 1.0f);
}

__global__ void k_dinv_inplace(float* __restrict__ deg, int N) {
    int i = blockIdx.x * blockDim.x + threadIdx.x;
    if (i < N) {
        float d = deg[i];
        deg[i] = (d > 0.0f) ? (1.0f / sqrtf(d)) : 0.0f;
    }
}

__global__ void k_dn_de_accum(const int* __restrict__ nidx, const int* __restrict__ eidx,
                              float* __restrict__ Dn, float* __restrict__ De, int nnz) {
    int i = blockIdx.x * blockDim.x + threadIdx.x;
    if (i < nnz) {
        atomicAdd(&Dn[nidx[i]], 1.0f);
        atomicAdd(&De[eidx[i]], 1.0f);
    }
}

// first nnz-index of each (sorted) hyperedge segment
__global__ void k_seg_start(const int* __restrict__ eidx, int* __restrict__ start, int nnz) {
    int i = blockIdx.x * blockDim.x + threadIdx.x;
    if (i < nnz) {
        if (i == 0 || eidx[i] != eidx[i - 1]) start[eidx[i]] = i;
    }
}

// ---------------------------------------------------------------- Laplacian scatter
// out[col[e], :] += scale * (-dinv[row[e]]*dinv[col[e]]) * v[row[e], :]
// one wave (32 lanes) per edge, float4 per lane
__global__ void k_lap_scatter(const float* __restrict__ v,
                              const int* __restrict__ row, const int* __restrict__ col,
                              const float* __restrict__ dinv,
                              float* __restrict__ out, float scale, int nE) {
    long tid = (long)blockIdx.x * blockDim.x + threadIdx.x;
    int e = (int)(tid >> 5);
    if (e >= nE) return;
    int f = ((int)tid & 31) * 4;
    int r = row[e], c = col[e];
    float w = -dinv[r] * dinv[c] * scale;
    float4 vv = *(const float4*)(v + (size_t)r * DH + f);
    float* o = out + (size_t)c * DH + f;
    atomicAdd(o + 0, w * vv.x);
    atomicAdd(o + 1, w * vv.y);
    atomicAdd(o + 2, w * vv.z);
    atomicAdd(o + 3, w * vv.w);
}

// ---------------------------------------------------------------- WMMA GEMM
// Register-blocked: ONE wave computes a full 16x128 slab of
//   C = sum_i A_i[M,128] @ W_i[128,128] (+bias)
// i.e. 8 column tiles held in 8 accumulators (64 VGPRs). Each A float2 load
// feeds 8 V_WMMA_F32_16X16X4_F32 ops (exact fp32), K-loop = 32 steps.
__device__ __forceinline__ void wmma_term8(v8f c[8], const float* __restrict__ A,
                                           const float* __restrict__ W,
                                           int m0, int lr, int half) {
    const int arow = (m0 + lr) * DH;
#pragma unroll 4
    for (int k0 = 0; k0 < DH; k0 += 4) {
        // A 16x4: lanes 0-15 -> K = k0,k0+1 ; lanes 16-31 -> K = k0+2,k0+3 (row M = lr)
        float2 av = *(const float2*)(A + arow + k0 + 2 * half);
        v2f a; a.x = av.x; a.y = av.y;
        // B 4x16: lanes 0-15 -> K = k0,k0+1 ; lanes 16-31 -> K = k0+2,k0+3 (col N = lr)
        const float* wr0 = W + (k0 + 2 * half) * DH + lr;
        const float* wr1 = wr0 + DH;
#pragma unroll
        for (int nt = 0; nt < 8; ++nt) {
            v2f b; b.x = wr0[nt * 16]; b.y = wr1[nt * 16];
            c[nt] = __builtin_amdgcn_wmma_f32_16x16x4_f32(false, a, false, b, (short)0,
                                                          c[nt], false, false);
        }
    }
}

__global__ void k_wmma_mm3(const float* __restrict__ A0, const float* __restrict__ W0,
                           const float* __restrict__ A1, const float* __restrict__ W1,
                           const float* __restrict__ A2, const float* __restrict__ W2,
                           const float* __restrict__ bias, float* __restrict__ C, int M) {
    const int lane = threadIdx.x & 31;
    const int wid  = (int)((blockIdx.x * (unsigned)blockDim.x + threadIdx.x) >> 5);
    const int lr   = lane & 15;
    const int half = lane >> 4;
    const int m0   = wid * 16;
    if (m0 >= M) return;                 // wave-uniform: EXEC stays all-1 for WMMA

    v8f c[8];
#pragma unroll
    for (int nt = 0; nt < 8; ++nt) {
        float bv = bias ? bias[nt * 16 + lr] : 0.0f;
#pragma unroll
        for (int r = 0; r < 8; ++r) c[nt][r] = bv;
    }

    wmma_term8(c, A0, W0, m0, lr, half);
    if (A1) wmma_term8(c, A1, W1, m0, lr, half);
    if (A2) wmma_term8(c, A2, W2, m0, lr, half);

    // C/D layout: VGPR r, lanes0-15 -> M = r ; lanes16-31 -> M = 8 + r
#pragma unroll
    for (int nt = 0; nt < 8; ++nt)
#pragma unroll
        for (int r = 0; r < 8; ++r)
            C[(size_t)(m0 + r + 8 * half) * DH + nt * 16 + lr] = c[nt][r];
}

// ---------------------------------------------------------------- hyperedge aggregation
// e[j,f] = PReLU( De_inv[j] * sum_rel (h[nidx[start+rel],f] + PE(rel,f)) + bias[f] )
__global__ void k_n2e_agg(const float* __restrict__ h, const int* __restrict__ nidx,
                          const int* __restrict__ start, const float* __restrict__ De,
                          const float* __restrict__ bias, const float* __restrict__ prelu_a,
                          float* __restrict__ e_out) {
    const int j = blockIdx.x;
    const int f = threadIdx.x;           // 128 threads, one per feature
    const float cntf = De[j];
    const int cnt = (int)cntf;
    // sinusoidal PE: div = exp(-(f&~1) * ln(10000)/128)
    const float divf = __expf(-(float)(f & ~1) * (9.210340371976184f / (float)DH));
    float acc = 0.0f;
    if (cnt > 0) {
        const int s = start[j];
        for (int rel = 0; rel < cnt; ++rel) {
            int nid = nidx[s + rel];
            float arg = (float)rel * divf;
            float pe = (f & 1) ? __cosf(arg) : __sinf(arg);
            acc += h[(size_t)nid * DH + f] + pe;
        }
    }
    float norm = (cnt > 0) ? (1.0f / cntf) : 0.0f;
    float ev = acc * norm + bias[f];
    float a = prelu_a[0];
    e_out[(size_t)j * DH + f] = (ev >= 0.0f) ? ev : a * ev;
}

// ---------------------------------------------------------------- edge->node scatter
// out[nidx[i], :] += Dn_inv[nidx[i]] * h2[eidx[i], :]
__global__ void k_e2n_scatter(const float* __restrict__ h2, const int* __restrict__ nidx,
                              const int* __restrict__ eidx, const float* __restrict__ Dn,
                              float* __restrict__ out, int nnz) {
    long tid = (long)blockIdx.x * blockDim.x + threadIdx.x;
    int i = (int)(tid >> 5);
    if (i >= nnz) return;
    int f = ((int)tid & 31) * 4;
    int nid = nidx[i], ej = eidx[i];
    float dn = Dn[nid];
    float w = (dn > 0.0f) ? (1.0f / dn) : 0.0f;
    float4 hv = *(const float4*)(h2 + (size_t)ej * DH + f);
    float* o = out + (size_t)nid * DH + f;
    atomicAdd(o + 0, w * hv.x);
    atomicAdd(o + 1, w * hv.y);
    atomicAdd(o + 2, w * hv.z);
    atomicAdd(o + 3, w * hv.w);
}

// ---------------------------------------------------------------- launch
extern "C" void kernel_launch(void* const* d_in, const int* in_sizes, int n_in,
                              void* d_out, int out_size, void* d_ws, size_t ws_size,
                              hipStream_t stream) {
    const float* x        = (const float*)d_in[0];
    const int*   eix      = (const int*)d_in[1];   // [2, nE]
    const int*   hix      = (const int*)d_in[2];   // [2, nnz]
    const float* cheb_W   = (const float*)d_in[3]; // [3,128,128]
    const float* cheb_b   = (const float*)d_in[4];
    const float* W_n2e    = (const float*)d_in[5];
    const float* W_e2n    = (const float*)d_in[6];
    const float* bias_n2e = (const float*)d_in[7];
    const float* bias_e2n = (const float*)d_in[8];
    const float* prelu_a  = (const float*)d_in[9];

    const int N   = in_sizes[0] / DH;       // 100000 (multiple of 16)
    const int nE  = in_sizes[1] / 2;        // 1600000
    const int nnz = in_sizes[2] / 2;        // 1000000
    const int Eh  = out_size / DH - N;      // 10000  (multiple of 16)

    const int* row  = eix;                  // edge_index[0]
    const int* col  = eix + nE;             // edge_index[1]
    const int* nidx = hix;                  // hyperedge_index[0]
    const int* eidx = hix + nnz;            // hyperedge_index[1]

    // ---- workspace layout ----
    float* Tx1   = (float*)d_ws;                    // N*DH (later reused as h)
    float* Tx2   = Tx1 + (size_t)N * DH;            // N*DH
    float* deg   = Tx2 + (size_t)N * DH;            // N   (becomes dinv in place)
    float* Dn    = deg + N;                         // N
    float* De    = Dn + N;                          // Eh
    int*   segst = (int*)(De + Eh);                 // Eh
    float* h2    = (float*)(segst + Eh);            // Eh*DH

    float* out_n = (float*)d_out;                   // [N,128]   x_low + n
    float* out_e = out_n + (size_t)N * DH;          // [Eh,128]  e

    const long nNF = (long)N * DH;
    // GEMM grids: one wave per 16-row slab, 8 waves per block
    const int gN  = (N / 16 + 7) / 8;
    const int gEh = (Eh / 16 + 7) / 8;

    // 1. zero accumulators
    k_zero_f32<<<(int)((nNF + TB - 1) / TB), TB, 0, stream>>>(Tx1, nNF);
    k_zero_f32<<<(N + TB - 1) / TB, TB, 0, stream>>>(deg, N);
    k_zero_f32<<<(N + TB - 1) / TB, TB, 0, stream>>>(Dn, N);
    k_zero_f32<<<(Eh + TB - 1) / TB, TB, 0, stream>>>(De, Eh);

    // 2. histograms / segment starts
    k_deg_accum<<<(nE + TB - 1) / TB, TB, 0, stream>>>(row, deg, nE);
    k_dn_de_accum<<<(nnz + TB - 1) / TB, TB, 0, stream>>>(nidx, eidx, Dn, De, nnz);
    k_seg_start<<<(nnz + TB - 1) / TB, TB, 0, stream>>>(eidx, segst, nnz);
    k_dinv_inplace<<<(N + TB - 1) / TB, TB, 0, stream>>>(deg, N);   // deg -> D^-1/2

    // 3. Chebyshev recurrence: Tx1 = L_hat x ; Tx2 = 2*L_hat*Tx1 - x
    {
        long thr = (long)nE * 32;
        k_lap_scatter<<<(int)((thr + TB - 1) / TB), TB, 0, stream>>>(
            x, row, col, deg, Tx1, 1.0f, nE);
        k_neg_copy<<<(int)((nNF + TB - 1) / TB), TB, 0, stream>>>(x, Tx2, nNF);
        k_lap_scatter<<<(int)((thr + TB - 1) / TB), TB, 0, stream>>>(
            Tx1, row, col, deg, Tx2, 2.0f, nE);
    }

    // 4. x_low = x@W0 + Tx1@W1 + Tx2@W2 + cheb_b  -> d_out node region
    k_wmma_mm3<<<gN, TB, 0, stream>>>(x, cheb_W,
                                      Tx1, cheb_W + DH * DH,
                                      Tx2, cheb_W + 2 * DH * DH,
                                      cheb_b, out_n, N);

    // 5. h = x_low @ W_n2e  (reuse Tx1 buffer)
    k_wmma_mm3<<<gN, TB, 0, stream>>>(out_n, W_n2e,
                                      nullptr, nullptr, nullptr, nullptr,
                                      nullptr, Tx1, N);

    // 6. hyperedge aggregation (+PE, +bias_n2e, PReLU) -> d_out e region
    k_n2e_agg<<<Eh, DH, 0, stream>>>(Tx1, nidx, segst, De, bias_n2e, prelu_a, out_e);

    // 7. h2 = e @ W_e2n
    k_wmma_mm3<<<gEh, TB, 0, stream>>>(out_e, W_e2n,
                                       nullptr, nullptr, nullptr, nullptr,
                                       nullptr, h2, Eh);

    // 8. node region: += bias_e2n, then scatter hyperedge messages back
    k_add_row_bias<<<(int)((nNF + TB - 1) / TB), TB, 0, stream>>>(out_n, bias_e2n, nNF);
    {
        long thr = (long)nnz * 32;
        k_e2n_scatter<<<(int)((thr + TB - 1) / TB), TB, 0, stream>>>(
            h2, nidx, eidx, Dn, out_n, nnz);
    }
}